// ResidualVectorQuantizer_43190191129293
// MI455X (gfx1250) — compile-verified
//
#include <hip/hip_runtime.h>
#include <hip/hip_bf16.h>

// ---------------------------------------------------------------------------
// Residual VQ (4 layers, Sinkhorn on layers 1..3) for MI455X / gfx1250.
//   - distance GEMM in full f32 via V_WMMA_F32_16X16X4_F32, 2x2 register
//     blocking per wave, double-buffered GLOBAL_LOAD_ASYNC_TO_LDS_B128
//     staging (ASYNCcnt-tracked, done-in-order => s_wait_asynccnt pipelining)
//   - Sinkhorn in dual-potential form: per iteration only
//     c_j = LSE_i(A - r_i), r_i = LSE_j(A - c_j). A (32MB) stays L2-resident.
//   - argmax_j(A - c_j) gives indices; gather + residual update + fused loss.
// ---------------------------------------------------------------------------

typedef __attribute__((ext_vector_type(2))) float v2f;
typedef __attribute__((ext_vector_type(8))) float v8f;

#define RVQ_N      8192
#define RVQ_E      768
#define RVQ_K      1024
#define RVQ_NE     (RVQ_N * RVQ_E)          // 6291456
#define RVQ_NK     (RVQ_N * RVQ_K)          // 8388608

// ---------------- init: residual = x, zero x_q output, zero loss ----------
__global__ __launch_bounds__(256) void rvq_init(const float* __restrict__ x,
                                                float* __restrict__ res,
                                                float* __restrict__ xq_out,
                                                float* __restrict__ loss_acc) {
    int i = blockIdx.x * 256 + threadIdx.x;
    if (i < RVQ_NE) {
        res[i]    = x[i];
        xq_out[i] = 0.0f;
    }
    if (i == 0) *loss_acc = 0.0f;
}

__global__ __launch_bounds__(256) void rvq_zero(float* __restrict__ p, int n) {
    int i = blockIdx.x * 256 + threadIdx.x;
    if (i < n) p[i] = 0.0f;
}

// ---------------- row squared-norms (xnorm / cnorm) -----------------------
__global__ __launch_bounds__(256) void rvq_rownorm(const float* __restrict__ src,
                                                   float* __restrict__ out,
                                                   int cols) {
    __shared__ float sb[256];
    const int row = blockIdx.x;
    const int tid = threadIdx.x;
    float s = 0.0f;
    const float* p = src + (size_t)row * cols;
    for (int e = tid; e < cols; e += 256) { float v = p[e]; s += v * v; }
    sb[tid] = s; __syncthreads();
    for (int off = 128; off > 0; off >>= 1) {
        if (tid < off) sb[tid] += sb[tid + off];
        __syncthreads();
    }
    if (tid == 0) out[row] = sb[0];
}

// ---------------- distance GEMM via WMMA f32 16x16x4 ----------------------
// D[i,k] = scale * ( xn[i] + en[k] - 2 * dot(res_i, cb_k) )
// Block tile 64(M) x 128(N), K-chunk 32, 256 threads = 8 waves (2Mx4N),
// each wave owns a 2x2 grid of 16x16 f32 accumulators (4 x v8f).
// Staging: GLOBAL_LOAD_ASYNC_TO_LDS_B128, double-buffered; async "done"
// returns in order, so s_wait_asynccnt <= (#next-chunk issues) releases the
// current chunk while the next streams in.
#define GT_M   64
#define GT_N   128
#define GT_KC  32
#define GT_STR 36                           // padded row stride (dwords): 144B,
                                            // 16B-aligned, 36*l % 64 distinct banks
#define GT_AS_DW   (GT_M * GT_STR)          // 2304 dwords
#define GT_BS_DW   (GT_N * GT_STR)          // 4608 dwords
#define GT_BUF_DW  (GT_AS_DW + GT_BS_DW)    // 6912 dwords = 27648 B
#define GT_SMEM_BYTES (2 * GT_BUF_DW * 4)   // 55296 B (double buffer)
#define GT_NCHUNK (RVQ_E / GT_KC)           // 24

__device__ __forceinline__ void rvq_async_b128(unsigned lds_byte,
                                               const float* gptr) {
    // VDST VGPR = wave-relative LDS byte address; VADDR = 64-bit global addr.
    asm volatile("global_load_async_to_lds_b128 %0, %1, off"
                 :: "v"(lds_byte), "v"(gptr) : "memory");
}

__global__ __launch_bounds__(256) void rvq_dist_gemm(const float* __restrict__ X,
                                                     const float* __restrict__ E,
                                                     const float* __restrict__ xn,
                                                     const float* __restrict__ en,
                                                     float* __restrict__ D,
                                                     float scale) {
    extern __shared__ float smem[];   // [2][As 64x36 | Bs 128x36]

    const int m0   = blockIdx.y * GT_M;
    const int n0   = blockIdx.x * GT_N;
    const int tid  = threadIdx.x;
    const int wave = tid >> 5;
    const int lane = tid & 31;
    const int half = lane >> 4;           // lane half selects K pair (ISA 7.12.2)
    const int l    = lane & 15;
    const int wmb  = (wave >> 2) * 32;    // wave M offset: 0 / 32
    const int wnb  = (wave & 3) * 32;     // wave N offset: 0 / 32 / 64 / 96

    v8f acc00 = {}, acc01 = {}, acc10 = {}, acc11 = {};

    // stage chunk `c` into buffer `buf` (6 async b128 per thread)
    auto stage = [&](int c, int buf) {
        const int k0 = c * GT_KC;
        const unsigned base = (unsigned)(buf * GT_BUF_DW * 4);
        #pragma unroll
        for (int i = 0; i < 2; ++i) {        // A: 64x32 dwords = 512 groups
            int g  = tid + i * 256;
            int m  = g >> 3;
            int k4 = (g & 7) * 4;
            rvq_async_b128(base + (unsigned)(m * GT_STR + k4) * 4,
                           X + (size_t)(m0 + m) * RVQ_E + k0 + k4);
        }
        #pragma unroll
        for (int i = 0; i < 4; ++i) {        // B: 128x32 dwords = 1024 groups
            int g  = tid + i * 256;
            int n  = g >> 3;
            int k4 = (g & 7) * 4;
            rvq_async_b128(base + (unsigned)(GT_AS_DW * 4) +
                               (unsigned)(n * GT_STR + k4) * 4,
                           E + (size_t)(n0 + n) * RVQ_E + k0 + k4);
        }
    };

    stage(0, 0);
    for (int c = 0; c < GT_NCHUNK; ++c) {
        const int cur = c & 1;
        if (c + 1 < GT_NCHUNK) {
            stage(c + 1, cur ^ 1);
            asm volatile("s_wait_asynccnt 0x6" ::: "memory");  // chunk c landed
        } else {
            asm volatile("s_wait_asynccnt 0x0" ::: "memory");
        }
        __syncthreads();

        const float* As = smem + cur * GT_BUF_DW;
        const float* Bs = As + GT_AS_DW;
        const float* Ar0 = As + (wmb + l) * GT_STR;
        const float* Ar1 = As + (wmb + 16 + l) * GT_STR;
        const float* Br0 = Bs + (wnb + l) * GT_STR;
        const float* Br1 = Bs + (wnb + 16 + l) * GT_STR;

        #pragma unroll
        for (int kk = 0; kk < GT_KC; kk += 4) {
            const int kh = kk + 2 * half;    // VGPR0=K0|K2, VGPR1=K1|K3
            v2f a0, a1, b0, b1;
            a0.x = Ar0[kh]; a0.y = Ar0[kh + 1];
            a1.x = Ar1[kh]; a1.y = Ar1[kh + 1];
            b0.x = Br0[kh]; b0.y = Br0[kh + 1];
            b1.x = Br1[kh]; b1.y = Br1[kh + 1];
            acc00 = __builtin_amdgcn_wmma_f32_16x16x4_f32(false, a0, false, b0,
                                                          (short)0, acc00, false, false);
            acc01 = __builtin_amdgcn_wmma_f32_16x16x4_f32(false, a0, false, b1,
                                                          (short)0, acc01, false, false);
            acc10 = __builtin_amdgcn_wmma_f32_16x16x4_f32(false, a1, false, b0,
                                                          (short)0, acc10, false, false);
            acc11 = __builtin_amdgcn_wmma_f32_16x16x4_f32(false, a1, false, b1,
                                                          (short)0, acc11, false, false);
        }
        __syncthreads();   // all waves done reading before buffer is re-staged
    }

    // C/D layout: VGPR v -> M = v + 8*half, N = lane&15
    #pragma unroll
    for (int tm = 0; tm < 2; ++tm) {
        #pragma unroll
        for (int tn = 0; tn < 2; ++tn) {
            const v8f& a = tm == 0 ? (tn == 0 ? acc00 : acc01)
                                   : (tn == 0 ? acc10 : acc11);
            #pragma unroll
            for (int v = 0; v < 8; ++v) {
                int row = m0 + wmb + tm * 16 + v + half * 8;
                int col = n0 + wnb + tn * 16 + l;
                float dv = xn[row] + en[col] - 2.0f * a[v];
                D[(size_t)row * RVQ_K + col] = dv * scale;
            }
        }
    }
}

// ---------------- online LSE combine --------------------------------------
__device__ __forceinline__ void lse_merge(float& m, float& s, float m2, float s2) {
    float M = fmaxf(m, m2);
    s = s * __expf(m - M) + s2 * __expf(m2 - M);
    m = M;
}

// c_j = logsumexp_i ( A[i,j] - r[i] ).  64 blocks x 256 threads.
// thread -> (col = blk*16 + tid&15, row-chunk = tid>>4 of 512 rows)
__global__ __launch_bounds__(256) void rvq_col_lse(const float* __restrict__ A,
                                                   const float* __restrict__ r,
                                                   float* __restrict__ c) {
    __shared__ float sm[256], ss[256];
    const int tid   = threadIdx.x;
    const int col   = blockIdx.x * 16 + (tid & 15);
    const int chunk = tid >> 4;
    const int row0  = chunk * (RVQ_N / 16);

    float m = -__builtin_inff(), s = 0.0f;
    for (int i = row0; i < row0 + (RVQ_N / 16); ++i) {
        float v = A[(size_t)i * RVQ_K + col] - r[i];
        float M = fmaxf(m, v);
        s = s * __expf(m - M) + __expf(v - M);
        m = M;
    }
    sm[tid] = m; ss[tid] = s; __syncthreads();
    for (int off = 128; off >= 16; off >>= 1) {
        if (tid < off) lse_merge(sm[tid], ss[tid], sm[tid + off], ss[tid + off]);
        __syncthreads();
    }
    if (tid < 16) c[blockIdx.x * 16 + tid] = sm[tid] + __logf(ss[tid]);
}

// r_i = logsumexp_j ( A[i,j] - c[j] ).  Wave per row, 1024 blocks x 256.
__global__ __launch_bounds__(256) void rvq_row_lse(const float* __restrict__ A,
                                                   const float* __restrict__ c,
                                                   float* __restrict__ r) {
    const int lane = threadIdx.x & 31;
    const int row  = blockIdx.x * 8 + (threadIdx.x >> 5);
    const float* Ar = A + (size_t)row * RVQ_K;

    float m = -__builtin_inff(), s = 0.0f;
    for (int j = lane; j < RVQ_K; j += 32) {
        float v = Ar[j] - c[j];
        float M = fmaxf(m, v);
        s = s * __expf(m - M) + __expf(v - M);
        m = M;
    }
    #pragma unroll
    for (int off = 16; off > 0; off >>= 1) {
        float m2 = __shfl_xor(m, off, 32);
        float s2 = __shfl_xor(s, off, 32);
        lse_merge(m, s, m2, s2);
    }
    if (lane == 0) r[row] = m + __logf(s);
}

// argmax_j ( sign * (A[i,j] - c?) ).  Wave per row.
__global__ __launch_bounds__(256) void rvq_arg_extreme(const float* __restrict__ A,
                                                       const float* __restrict__ c,
                                                       int* __restrict__ idx,
                                                       int use_c, int maximize) {
    const int lane = threadIdx.x & 31;
    const int row  = blockIdx.x * 8 + (threadIdx.x >> 5);
    const float* Ar = A + (size_t)row * RVQ_K;
    const float sgn = maximize ? 1.0f : -1.0f;

    float best = -__builtin_inff();
    int bidx = 0;
    for (int j = lane; j < RVQ_K; j += 32) {
        float v = Ar[j] - (use_c ? c[j] : 0.0f);
        v *= sgn;
        if (v > best) { best = v; bidx = j; }
    }
    #pragma unroll
    for (int off = 16; off > 0; off >>= 1) {
        float v2 = __shfl_xor(best, off, 32);
        int   i2 = __shfl_xor(bidx, off, 32);
        if (v2 > best || (v2 == best && i2 < bidx)) { best = v2; bidx = i2; }
    }
    if (lane == 0) idx[row] = bidx;
}

// gather codebook row, accumulate x_q, update residual, fused loss sums.
__global__ __launch_bounds__(256) void rvq_gather_update(const float* __restrict__ cb,
                                                         const int* __restrict__ idx,
                                                         float* __restrict__ res,
                                                         float* __restrict__ xq_out,
                                                         float* __restrict__ loss_acc,
                                                         float* __restrict__ idx_out,
                                                         int layer) {
    __shared__ float sb[256];
    const int n   = blockIdx.x;
    const int tid = threadIdx.x;
    const int k   = idx[n];
    const float* cbr = cb + (size_t)k * RVQ_E;

    float local = 0.0f;
    for (int e = tid; e < RVQ_E; e += 256) {
        size_t o = (size_t)n * RVQ_E + e;
        float rv = res[o];
        float qv = cbr[e];
        float d  = qv - rv;
        local += d * d;
        res[o]    = rv - qv;      // residual for next layer
        xq_out[o] += qv;          // forward value of STE path is the code row
    }
    sb[tid] = local; __syncthreads();
    for (int off = 128; off > 0; off >>= 1) {
        if (tid < off) sb[tid] += sb[tid + off];
        __syncthreads();
    }
    if (tid == 0) {
        atomicAdd(loss_acc, sb[0]);
        idx_out[(size_t)n * 4 + layer] = (float)k;
    }
}

__global__ void rvq_finalize_loss(const float* __restrict__ loss_acc,
                                  float* __restrict__ out_loss) {
    // mean over 4 layers of 2*MSE  ==  total_sq / (2*N*E)
    *out_loss = loss_acc[0] / (2.0f * (float)RVQ_NE);
}

// ---------------------------------------------------------------------------
extern "C" void kernel_launch(void* const* d_in, const int* in_sizes, int n_in,
                              void* d_out, int out_size, void* d_ws, size_t ws_size,
                              hipStream_t stream) {
    const float* x = (const float*)d_in[0];
    const float* cbs[4] = { (const float*)d_in[1], (const float*)d_in[2],
                            (const float*)d_in[3], (const float*)d_in[4] };

    float* out_xq   = (float*)d_out;                 // [8192*768]
    float* out_loss = out_xq + RVQ_NE;               // [1]
    float* out_idx  = out_loss + 1;                  // [8192*4] (as float)

    float* wsf  = (float*)d_ws;
    float* res  = wsf;                  wsf += RVQ_NE;    // residual
    float* A    = wsf;                  wsf += RVQ_NK;    // dist / logQ base
    float* xn   = wsf;                  wsf += RVQ_N;
    float* cn   = wsf;                  wsf += RVQ_K;
    float* rr   = wsf;                  wsf += RVQ_N;     // row potential
    float* cc   = wsf;                  wsf += RVQ_K;     // col potential
    float* lacc = wsf;                  wsf += 1;
    int*   idx  = (int*)wsf;

    const float SK_EPS[4] = { 0.0f, 0.003f, 0.003f, 0.003f };
    const int   SK_ITERS  = 100;

    rvq_init<<<(RVQ_NE + 255) / 256, 256, 0, stream>>>(x, res, out_xq, lacc);

    for (int layer = 0; layer < 4; ++layer) {
        const float eps = SK_EPS[layer];
        const float scale = (eps > 0.0f) ? (-1.0f / eps) : 1.0f;

        rvq_rownorm<<<RVQ_N, 256, 0, stream>>>(res, xn, RVQ_E);
        rvq_rownorm<<<RVQ_K, 256, 0, stream>>>(cbs[layer], cn, RVQ_E);

        dim3 ggrid(RVQ_K / GT_N, RVQ_N / GT_M);   // (8, 128)
        rvq_dist_gemm<<<ggrid, 256, GT_SMEM_BYTES, stream>>>(
            res, cbs[layer], xn, cn, A, scale);

        if (eps > 0.0f) {
            rvq_zero<<<(RVQ_N + 255) / 256, 256, 0, stream>>>(rr, RVQ_N);
            for (int it = 0; it < SK_ITERS; ++it) {
                rvq_col_lse<<<RVQ_K / 16, 256, 0, stream>>>(A, rr, cc);
                rvq_row_lse<<<RVQ_N / 8, 256, 0, stream>>>(A, cc, rr);
            }
            rvq_arg_extreme<<<RVQ_N / 8, 256, 0, stream>>>(A, cc, idx, 1, 1);
        } else {
            rvq_arg_extreme<<<RVQ_N / 8, 256, 0, stream>>>(A, nullptr, idx, 0, 0);
        }

        rvq_gather_update<<<RVQ_N, 256, 0, stream>>>(cbs[layer], idx, res,
                                                     out_xq, lacc, out_idx, layer);
    }

    rvq_finalize_loss<<<1, 1, 0, stream>>>(lacc, out_loss);
}